// MyModel_61933428411506
// MI455X (gfx1250) — compile-verified
//
#include <hip/hip_runtime.h>

// out = S * Xp * S^T per 40x40 image (Xp zero-padded to 48x48),
// S = tridiagonal ones band  ==  3x3 ones depthwise conv, pad=1.
// One wave32 per image. Per 16x16 output tile we sweep a K=20 band window
// (5x v_wmma_f32_16x16x4_f32) covering rows/cols 16b-2..16b+17 with
// A[m][kl] = band(m, kl-2); the 2-element overhang has zero coefficients,
// so no scalar corner corrections (no cndmask chains) are needed.

typedef __attribute__((ext_vector_type(2))) float v2f;
typedef __attribute__((ext_vector_type(8))) float v8f;

#define IMG_H 40
#define IMG_W 40
#define IMG_N (IMG_H * IMG_W)

// LDS image Xp: 52 rows x 48 cols, image row r stored at row r+2
// (rows 0,1 and 42..51 are zero guards)
#define XR 52
#define XC 48
// LDS intermediate T = S*Xp: 48 rows x 52 cols, col c stored at col c+2
// (cols 0,1 and 50,51 are zero guards)
#define TR 48
#define TC 52
#define PERWAVE (XR * XC + TR * TC)   // 4992 floats per wave
#define WAVES_PER_BLOCK 2

__device__ __forceinline__ float band(int i, int j) {
  // 1.0 iff |i-j| <= 1
  return ((unsigned)(i - j + 1) <= 2u) ? 1.0f : 0.0f;
}

__global__ __launch_bounds__(WAVES_PER_BLOCK * 32)
void fold_sum3x3_wmma(const float* __restrict__ x, float* __restrict__ out, int nimg)
{
  __shared__ __align__(16) float lds[WAVES_PER_BLOCK * PERWAVE];
  const int lane = threadIdx.x & 31;
  const int wave = threadIdx.x >> 5;
  const int img  = blockIdx.x * WAVES_PER_BLOCK + wave;
  if (img >= nimg) return;                 // wave-uniform

  float* xs = lds + wave * PERWAVE;        // Xp  [XR][XC]
  float* ts = xs + XR * XC;                // T   [TR][TC]

  // ---- zero-init both wave-private LDS buffers (guards must be 0) ----
  {
    const float4 z = make_float4(0.f, 0.f, 0.f, 0.f);
    for (int i = lane * 4; i < PERWAVE; i += 32 * 4)
      *(float4*)(xs + i) = z;
  }

  // ---- stage image into LDS (coalesced float4), image row r -> LDS row r+2 ----
  {
    const float* src = x + (size_t)img * IMG_N;
    for (int i = lane; i < (IMG_H * IMG_W) / 4; i += 32) {   // 400 float4 segs
      int r = i / 10, s = i % 10;
      float4 v = *(const float4*)(src + r * IMG_W + s * 4);
      *(float4*)(xs + (r + 2) * XC + s * 4) = v;
    }
  }
  // same-wave DS ops are in order; no barrier needed (LDS slice is wave-private)

  const int m  = lane & 15;            // A-row / B-col / D-col index
  const int hk = (lane >> 4) & 1;      // K-half select for A/B operands
  const int h8 = hk ? 8 : 0;           // C/D row-half offset

  // ================= GEMM1: T = S * Xp (vertical 3-sum) =================
  // T rows 16bi+m = sum_kl band(m, kl-2) * Xp[16bi-2+kl][:],  kl = 0..19
  for (int bi = 0; bi < 3; ++bi) {
    for (int bj = 0; bj < 3; ++bj) {
      v8f acc = {0.f, 0.f, 0.f, 0.f, 0.f, 0.f, 0.f, 0.f};
#pragma unroll
      for (int kc = 0; kc < 5; ++kc) {
        const int k0 = 4 * kc + 2 * hk;                      // kl for vgpr0
        v2f a, b;
        a.x = band(m, k0 - 2);
        a.y = band(m, k0 - 1);
        // Xp row (16bi - 2 + kl) at LDS row (16bi + kl) after +2 shift
        b.x = xs[(16 * bi + k0)     * XC + 16 * bj + m];
        b.y = xs[(16 * bi + k0 + 1) * XC + 16 * bj + m];
        acc = __builtin_amdgcn_wmma_f32_16x16x4_f32(
                  false, a, false, b, (short)0, acc, false, false);
      }
#pragma unroll
      for (int v = 0; v < 8; ++v)
        ts[(16 * bi + v + h8) * TC + 16 * bj + m + 2] = acc[v];  // col +2 shift
    }
  }

  // ================= GEMM2: Out = T * S^T (horizontal 3-sum) =================
  // Out[:, 16bj+n] = sum_kl T[:, 16bj-2+kl] * band(n, kl-2),  kl = 0..19
  float* dst = out + (size_t)img * IMG_N;
  for (int bi = 0; bi < 3; ++bi) {
    for (int bj = 0; bj < 3; ++bj) {
      v8f acc = {0.f, 0.f, 0.f, 0.f, 0.f, 0.f, 0.f, 0.f};
#pragma unroll
      for (int kc = 0; kc < 5; ++kc) {
        const int k0 = 4 * kc + 2 * hk;                      // kl for vgpr0
        v2f a, b;
        // T col (16bj - 2 + kl) at LDS col (16bj + kl) after +2 shift
        a.x = ts[(16 * bi + m) * TC + 16 * bj + k0];
        a.y = ts[(16 * bi + m) * TC + 16 * bj + k0 + 1];
        b.x = band(m, k0 - 2);
        b.y = band(m, k0 - 1);
        acc = __builtin_amdgcn_wmma_f32_16x16x4_f32(
                  false, a, false, b, (short)0, acc, false, false);
      }
#pragma unroll
      for (int v = 0; v < 8; ++v) {
        const int r = 16 * bi + v + h8;
        const int c = 16 * bj + m;
        if (r < IMG_H && c < IMG_W) dst[r * IMG_W + c] = acc[v];
      }
    }
  }
}

extern "C" void kernel_launch(void* const* d_in, const int* in_sizes, int n_in,
                              void* d_out, int out_size, void* d_ws, size_t ws_size,
                              hipStream_t stream) {
  const float* x = (const float*)d_in[0];
  float* out = (float*)d_out;
  const int nimg = in_sizes[0] / IMG_N;                  // 64*512 = 32768
  const int blocks = (nimg + WAVES_PER_BLOCK - 1) / WAVES_PER_BLOCK;
  fold_sum3x3_wmma<<<blocks, WAVES_PER_BLOCK * 32, 0, stream>>>(x, out, nimg);
}